// LyraGemma3Attention_53498112639338
// MI455X (gfx1250) — compile-verified
//
#include <hip/hip_runtime.h>
#include <math.h>

// ---------------- problem constants ----------------
#define B_      2
#define S_      2048
#define HID_    2560
#define H_      8
#define KV_     4
#define D_      256
#define HD_     (H_ * D_)     // 2048
#define KVD_    (KV_ * D_)    // 1024
#define WINDOW_ 1024
#define SCALING_ 0.0625f      // 256^-0.5
#define EPS_    1e-6f
#define THETA_  10000.0f

typedef float v2f __attribute__((ext_vector_type(2)));
typedef float v8f __attribute__((ext_vector_type(8)));
typedef int   i4v __attribute__((vector_size(4 * sizeof(int))));

__device__ __forceinline__ v8f vzero8() {
  v8f z;
#pragma unroll
  for (int i = 0; i < 8; ++i) z[i] = 0.0f;
  return z;
}

// ---------------------------------------------------------------------
// Async global->LDS 16B copy (gfx1250 GLOBAL_LOAD_ASYNC_TO_LDS_B128,
// tracked by ASYNCcnt). Builtin signature (from hipcc diagnostic):
//   (int4 addrspace(1)*, int4 addrspace(3)*, imm offset, imm cpol)
// Falls back to a VGPR round trip if the builtin is not declared.
// ---------------------------------------------------------------------
#define AS1 __attribute__((address_space(1)))
#define AS3 __attribute__((address_space(3)))

__device__ __forceinline__ void async_copy16(const float* g, float* l) {
#if __has_builtin(__builtin_amdgcn_global_load_async_to_lds_b128)
  __builtin_amdgcn_global_load_async_to_lds_b128(
      (AS1 i4v*)g, (AS3 i4v*)l, 0, 0);
#else
  *(float4*)l = *(const float4*)g;
#endif
}

__device__ __forceinline__ void async_wait0() {
#if __has_builtin(__builtin_amdgcn_s_wait_asynccnt)
  __builtin_amdgcn_s_wait_asynccnt(0);
#elif __has_builtin(__builtin_amdgcn_global_load_async_to_lds_b128)
  asm volatile("s_wait_asynccnt 0x0" ::: "memory");
#endif
}

// =====================================================================
// Generic fp32 WMMA GEMM: C[M×N] = A[M×K] * B[K×N], all row-major.
// Block = 128 threads (4 waves). Block tile 64×64; wave w owns rows
// [w*16, w*16+16) × 64 cols (4 C tiles of 16×16). K staged in chunks of 16
// through LDS via the async global->LDS path.
// =====================================================================
#define AST 20   // A tile row stride: 80B rows (16B aligned), conflict-free
#define BST 72   // B tile row stride: 288B rows (16B aligned)

__global__ __launch_bounds__(128) void gemm_f32_wmma(
    const float* __restrict__ A, const float* __restrict__ Bm,
    float* __restrict__ C, int M, int N, int K) {
  __shared__ float sA[64 * AST];
  __shared__ float sB[16 * BST];

  const int tid  = threadIdx.x;
  const int lane = tid & 31;
  const int wave = tid >> 5;
  const int half = lane >> 4;   // 0 or 1 (wave32 halves)
  const int l16  = lane & 15;
  const int n0 = blockIdx.x * 64;
  const int m0 = blockIdx.y * 64;

  v8f acc[4];
#pragma unroll
  for (int t = 0; t < 4; ++t) acc[t] = vzero8();

  for (int k0 = 0; k0 < K; k0 += 16) {
    __syncthreads();   // previous chunk's fragment reads must be done
    // ---- stage A 64x16 (async, 2 x 16B per thread) ----
#pragma unroll
    for (int it = 0; it < 2; ++it) {
      int idx = tid + 128 * it;          // 0..255 float4s
      int r   = idx >> 2;                // 0..63
      int c4  = (idx & 3) * 4;           // 0,4,8,12
      async_copy16(A + (size_t)(m0 + r) * K + k0 + c4, &sA[r * AST + c4]);
    }
    // ---- stage B 16x64 (async, 2 x 16B per thread) ----
#pragma unroll
    for (int it = 0; it < 2; ++it) {
      int idx = tid + 128 * it;          // 0..255 float4s
      int r   = idx >> 4;                // 0..15
      int c4  = (idx & 15) * 4;          // 0..60
      async_copy16(Bm + (size_t)(k0 + r) * N + n0 + c4, &sB[r * BST + c4]);
    }
    async_wait0();
    __syncthreads();

    // ---- 16 WMMAs (4 k-substeps × 4 col tiles) ----
#pragma unroll
    for (int kk = 0; kk < 16; kk += 4) {
      v2f a;
      // A layout: lane<16 holds K=kk,kk+1 ; lane>=16 holds K=kk+2,kk+3
      a.x = sA[(wave * 16 + l16) * AST + kk + half * 2 + 0];
      a.y = sA[(wave * 16 + l16) * AST + kk + half * 2 + 1];
#pragma unroll
      for (int ct = 0; ct < 4; ++ct) {
        v2f b;
        b.x = sB[(kk + half * 2 + 0) * BST + ct * 16 + l16];
        b.y = sB[(kk + half * 2 + 1) * BST + ct * 16 + l16];
        acc[ct] = __builtin_amdgcn_wmma_f32_16x16x4_f32(
            false, a, false, b, (short)0, acc[ct], false, false);
      }
    }
  }

  // ---- write C (C layout: vgpr i -> row half*8+i, lane%16 -> col) ----
#pragma unroll
  for (int ct = 0; ct < 4; ++ct)
#pragma unroll
    for (int i = 0; i < 8; ++i) {
      int row = m0 + wave * 16 + half * 8 + i;
      int col = n0 + ct * 16 + l16;
      C[(size_t)row * N + col] = acc[ct][i];
    }
}

// =====================================================================
// RMS-norm + RoPE. One wave per (b, s, head-task).
// Tasks 0..7  : q head h   -> norm(q_norm_w) + rope, in place on qbuf.
// Tasks 8..11 : kv head    -> norm(k_norm_w): un-roped copy to knbuf,
//                             roped written in place on kbuf.
// Lane holds d = lane + 32*t (t=0..7); rope partner (d±128) is t±4 on
// the SAME lane, so no cross-lane traffic for rotate_half.
// =====================================================================
__global__ __launch_bounds__(256) void normrope_kernel(
    float* __restrict__ qbuf, float* __restrict__ kbuf,
    float* __restrict__ knbuf,
    const float* __restrict__ qw, const float* __restrict__ kw) {
  const int gwave = blockIdx.x * 8 + (threadIdx.x >> 5);
  const int lane  = threadIdx.x & 31;
  const int task  = gwave % 12;       // 0..11
  const int bs    = gwave / 12;       // 0..B*S-1
  if (bs >= B_ * S_) return;
  const int s = bs % S_;

  const bool is_q = (task < 8);
  float* base;
  const float* wvec;
  if (is_q) {
    base = qbuf + (size_t)bs * HD_ + task * D_;
    wvec = qw;
  } else {
    base = kbuf + (size_t)bs * KVD_ + (task - 8) * D_;
    wvec = kw;
  }

  float x[8];
  float ss = 0.0f;
#pragma unroll
  for (int t = 0; t < 8; ++t) {
    x[t] = base[lane + 32 * t];
    ss += x[t] * x[t];
  }
#pragma unroll
  for (int off = 1; off < 32; off <<= 1) ss += __shfl_xor(ss, off, 32);
  const float scale = rsqrtf(ss * (1.0f / (float)D_) + EPS_);

  float n[8];
#pragma unroll
  for (int t = 0; t < 8; ++t) {
    int d = lane + 32 * t;
    n[t] = x[t] * scale * (1.0f + wvec[d]);
  }

  if (!is_q) {  // store un-roped normed K (used by lyra heads)
    float* kn = knbuf + (size_t)bs * KVD_ + (task - 8) * D_;
#pragma unroll
    for (int t = 0; t < 8; ++t) kn[lane + 32 * t] = n[t];
  }

  const float pos = (float)s;
#pragma unroll
  for (int t = 0; t < 8; ++t) {
    int d = lane + 32 * t;
    int j = d & 127;  // emb = concat(freqs, freqs)
    float f   = pos * powf(THETA_, -(float)(2 * j) / (float)D_);
    float c   = cosf(f), si = sinf(f);
    float rot = (t < 4) ? -n[t + 4] : n[t - 4];
    base[d] = n[t] * c + rot * si;
  }
}

// =====================================================================
// Flash-style sliding-window attention. One wave (32 thr) per block.
// grid = (S/16, H, B). Heads 0..3 use roped K, 4..7 use un-roped normed K.
// Q tile 16×256 staged once in LDS (async); per 16-key block K/V staged
// (async b128), scores via WMMA f32 16x16x4 (64 per block), online
// softmax, P -> LDS (layout conversion C->A), P·V via 64 more WMMAs.
// =====================================================================
#define QSTR 260   // padded row stride: 1040B rows (16B aligned), 260%64==4
#define PSTR 17

__global__ __launch_bounds__(32) void attn_kernel(
    const float* __restrict__ qn, const float* __restrict__ kr,
    const float* __restrict__ kn, const float* __restrict__ v,
    float* __restrict__ o) {
  __shared__ float sQ[16 * QSTR];
  __shared__ float sK[16 * QSTR];
  __shared__ float sV[16 * QSTR];
  __shared__ float sP[16 * PSTR];

  const int lane = threadIdx.x;
  const int half = lane >> 4;
  const int l16  = lane & 15;
  const int qt = blockIdx.x;
  const int h  = blockIdx.y;
  const int b  = blockIdx.z;
  const int i0 = qt * 16;
  const int kv = (h < 4) ? h : (h - 4);
  const float* kbase = (h < 4) ? kr : kn;

  // ---- stage Q tile (16 rows × 256), async ----
#pragma unroll 4
  for (int it = 0; it < 32; ++it) {
    int idx = lane + 32 * it;          // 0..1023 float4s
    int r   = idx >> 6;                // 0..15
    int c4  = (idx & 63) * 4;          // 0..252
    async_copy16(qn + (size_t)(b * S_ + i0 + r) * HD_ + h * D_ + c4,
                 &sQ[r * QSTR + c4]);
  }
  async_wait0();
  __syncthreads();

  float m[8], lsum[8];
  v8f Oacc[16];
#pragma unroll
  for (int i = 0; i < 8; ++i) { m[i] = -1e30f; lsum[i] = 0.0f; }
#pragma unroll
  for (int c = 0; c < 16; ++c) Oacc[c] = vzero8();

  const int kb_lo = (i0 - WINDOW_ > 0) ? (i0 - WINDOW_) : 0;

  for (int kb = kb_lo; kb <= i0; kb += 16) {
    __syncthreads();   // prior fragment reads done before LDS overwrite
    // ---- stage K and V blocks (16 keys × 256), async ----
#pragma unroll 4
    for (int it = 0; it < 32; ++it) {
      int idx = lane + 32 * it;
      int r   = idx >> 6;
      int c4  = (idx & 63) * 4;
      size_t goff = (size_t)(b * S_ + kb + r) * KVD_ + kv * D_ + c4;
      async_copy16(kbase + goff, &sK[r * QSTR + c4]);
      async_copy16(v + goff, &sV[r * QSTR + c4]);
    }
    async_wait0();
    __syncthreads();

    // ---- scores = Q · K^T  (16×16 tile, 64 WMMAs over D=256) ----
    v8f sc = vzero8();
    for (int dk = 0; dk < D_; dk += 4) {
      v2f a, bf;
      a.x  = sQ[l16 * QSTR + dk + half * 2 + 0];
      a.y  = sQ[l16 * QSTR + dk + half * 2 + 1];
      bf.x = sK[l16 * QSTR + dk + half * 2 + 0];  // B[k=d][n=key] = K[key][d]
      bf.y = sK[l16 * QSTR + dk + half * 2 + 1];
      sc = __builtin_amdgcn_wmma_f32_16x16x4_f32(
          false, a, false, bf, (short)0, sc, false, false);
    }

    // ---- mask + online softmax (rows live across 16 lanes per vgpr) ----
#pragma unroll
    for (int i = 0; i < 8; ++i) {
      int iq = i0 + half * 8 + i;     // global query row
      int j  = kb + l16;              // global key col
      bool ok = (j <= iq) && (j > iq - WINDOW_);
      float sv = ok ? sc[i] * SCALING_ : -1e30f;
      float mx = sv;
#pragma unroll
      for (int off = 1; off < 16; off <<= 1) mx = fmaxf(mx, __shfl_xor(mx, off, 32));
      float mnew = fmaxf(m[i], mx);
      float p  = ok ? expf(sv - mnew) : 0.0f;
      float ps = p;
#pragma unroll
      for (int off = 1; off < 16; off <<= 1) ps += __shfl_xor(ps, off, 32);
      float alpha = expf(m[i] - mnew);
      lsum[i] = lsum[i] * alpha + ps;
      m[i] = mnew;
#pragma unroll
      for (int c = 0; c < 16; ++c) Oacc[c][i] *= alpha;
      sP[(half * 8 + i) * PSTR + l16] = p;   // P row-major for A-layout reads
    }
    __syncthreads();

    // ---- O += P · V  (16 col tiles × 4 k-substeps = 64 WMMAs) ----
#pragma unroll
    for (int c = 0; c < 16; ++c) {
#pragma unroll
      for (int kk = 0; kk < 16; kk += 4) {
        v2f a, bf;
        a.x  = sP[l16 * PSTR + kk + half * 2 + 0];
        a.y  = sP[l16 * PSTR + kk + half * 2 + 1];
        bf.x = sV[(kk + half * 2 + 0) * QSTR + c * 16 + l16];
        bf.y = sV[(kk + half * 2 + 1) * QSTR + c * 16 + l16];
        Oacc[c] = __builtin_amdgcn_wmma_f32_16x16x4_f32(
            false, a, false, bf, (short)0, Oacc[c], false, false);
      }
    }
  }

  // ---- normalize + write (b, s, h*D+d) ----
#pragma unroll
  for (int i = 0; i < 8; ++i) {
    float inv = 1.0f / lsum[i];
    int row = i0 + half * 8 + i;
#pragma unroll
    for (int c = 0; c < 16; ++c) {
      o[(size_t)(b * S_ + row) * HD_ + h * D_ + c * 16 + l16] = Oacc[c][i] * inv;
    }
  }
}

// =====================================================================
// Host-side launch
// =====================================================================
extern "C" void kernel_launch(void* const* d_in, const int* in_sizes, int n_in,
                              void* d_out, int out_size, void* d_ws, size_t ws_size,
                              hipStream_t stream) {
  (void)in_sizes; (void)n_in; (void)out_size; (void)ws_size;
  const float* hs = (const float*)d_in[0];   // (B,S,HID)
  const float* wq = (const float*)d_in[1];   // (HID, H*D)
  const float* wk = (const float*)d_in[2];   // (HID, KV*D)
  const float* wv = (const float*)d_in[3];   // (HID, KV*D)
  const float* wo = (const float*)d_in[4];   // (H*D, HID)
  const float* qw = (const float*)d_in[5];   // (D,)
  const float* kw = (const float*)d_in[6];   // (D,)
  float* out = (float*)d_out;

  float* ws    = (float*)d_ws;
  float* qbuf  = ws;                                 // BS × HD   (8 M)
  float* kbuf  = qbuf  + (size_t)B_ * S_ * HD_;      // BS × KVD  (4 M)
  float* vbuf  = kbuf  + (size_t)B_ * S_ * KVD_;     // BS × KVD  (4 M)
  float* knbuf = vbuf  + (size_t)B_ * S_ * KVD_;     // BS × KVD  (4 M)
  float* abuf  = knbuf + (size_t)B_ * S_ * KVD_;     // BS × HD   (8 M)

  const int M = B_ * S_;   // 4096

  // QKV projections
  gemm_f32_wmma<<<dim3(HD_ / 64, M / 64), 128, 0, stream>>>(hs, wq, qbuf, M, HD_, HID_);
  gemm_f32_wmma<<<dim3(KVD_ / 64, M / 64), 128, 0, stream>>>(hs, wk, kbuf, M, KVD_, HID_);
  gemm_f32_wmma<<<dim3(KVD_ / 64, M / 64), 128, 0, stream>>>(hs, wv, vbuf, M, KVD_, HID_);

  // RMS norm + RoPE (q in-place; k -> knbuf (no rope) and in-place rope)
  normrope_kernel<<<(M * 12) / 8, 256, 0, stream>>>(qbuf, kbuf, knbuf, qw, kw);

  // Sliding-window attention
  attn_kernel<<<dim3(S_ / 16, H_, B_), 32, 0, stream>>>(qbuf, kbuf, knbuf, vbuf, abuf);

  // Output projection
  gemm_f32_wmma<<<dim3(HID_ / 64, M / 64), 128, 0, stream>>>(abuf, wo, out, M, HID_, HD_);
}